// ApplyLUT_85521388798495
// MI455X (gfx1250) — compile-verified
//
#include <hip/hip_runtime.h>
#include <stdint.h>

// ---------------------------------------------------------------------------
// 3D LUT (33^3, trilinear) applied to a (4,3,2048,2048) fp32 NCHW image.
// Bandwidth-bound: ~402 MB streamed -> ~17us floor @ 23.3 TB/s.
// Strategy:
//  * Repack LUT entries (12B) -> padded float4 (16B) in d_ws so every corner
//    fetch is one aligned global_load_b128 (8 gathers/pixel, pairs along b
//    are contiguous because idx1 == idx0+1 always).
//  * Stream R/G/B image planes through LDS with CDNA5 async copies
//    (global_load_async_to_lds_b128, th:TH_LOAD_NT so the 384MB stream does
//    not evict the hot LUT from WGP$/L2), double-buffered per wave, synced
//    with s_wait_asynccnt (no barriers: each wave consumes only its own
//    async loads, which complete in order).
//  * Non-temporal stores for the output for the same cache-residency reason.
// ---------------------------------------------------------------------------

typedef float v4f __attribute__((ext_vector_type(4)));

#define LUTN   33
#define LUTENT (33 * 33 * 33)   // 35937 entries

constexpr int      BLOCK       = 256;
constexpr int      ITER        = 8;                 // chunks per block
constexpr unsigned PLANE_GRP   = (2048u * 2048u) / 4u;  // 2^20 float4-groups per plane

// ---- CDNA5 async global->LDS copy (16B per lane), non-temporal ------------
__device__ __forceinline__ void async_copy16(const v4f* gsrc, v4f* lds_dst) {
    // LDS byte address = low 32 bits of the generic shared-memory address.
    unsigned lds_addr = (unsigned)(uintptr_t)lds_dst;
    asm volatile("global_load_async_to_lds_b128 %0, %1, off th:TH_LOAD_NT"
                 :
                 : "v"(lds_addr), "v"(gsrc)
                 : "memory");
}

__device__ __forceinline__ void wait_async_le3() {
    asm volatile("s_wait_asynccnt 3" ::: "memory");
}
__device__ __forceinline__ void wait_async_0() {
    asm volatile("s_wait_asynccnt 0" ::: "memory");
}

// ---- trilinear helpers -----------------------------------------------------
struct F3 { float x, y, z; };

__device__ __forceinline__ F3 mix3(F3 a, F3 b, float f) {
    float omf = 1.0f - f;
    F3 r;
    r.x = a.x * omf + b.x * f;
    r.y = a.y * omf + b.y * f;
    r.z = a.z * omf + b.z * f;
    return r;
}

template <bool PADDED>
__device__ __forceinline__ F3 sample_lut(const v4f* __restrict__ lutp,
                                         const float* __restrict__ lut,
                                         float r, float g, float b) {
    float sr = r * 32.0f, sg = g * 32.0f, sb = b * 32.0f;
    float wr = floorf(sr), wg = floorf(sg), wb = floorf(sb);
    float tr = sr - wr, tg = sg - wg, tb = sb - wb;   // unclamped frac (ref)
    int ir = (int)wr, ig = (int)wg, ib = (int)wb;
    ir = min(max(ir, 0), LUTN - 2);
    ig = min(max(ig, 0), LUTN - 2);
    ib = min(max(ib, 0), LUTN - 2);
    // entry index; +1 along b, +33 along g, +1089 along r
    int e = ir * (LUTN * LUTN) + ig * LUTN + ib;

    F3 c000, c001, c010, c011, c100, c101, c110, c111;
    if constexpr (PADDED) {
        v4f q;
        q = lutp[e];                 c000 = {q.x, q.y, q.z};
        q = lutp[e + 1];             c001 = {q.x, q.y, q.z};
        q = lutp[e + LUTN];          c010 = {q.x, q.y, q.z};
        q = lutp[e + LUTN + 1];      c011 = {q.x, q.y, q.z};
        q = lutp[e + LUTN * LUTN];       c100 = {q.x, q.y, q.z};
        q = lutp[e + LUTN * LUTN + 1];   c101 = {q.x, q.y, q.z};
        q = lutp[e + LUTN * LUTN + LUTN];     c110 = {q.x, q.y, q.z};
        q = lutp[e + LUTN * LUTN + LUTN + 1]; c111 = {q.x, q.y, q.z};
    } else {
        const float* p = lut + (size_t)e * 3;
        c000 = {p[0], p[1], p[2]};  c001 = {p[3], p[4], p[5]};
        p = lut + (size_t)(e + LUTN) * 3;
        c010 = {p[0], p[1], p[2]};  c011 = {p[3], p[4], p[5]};
        p = lut + (size_t)(e + LUTN * LUTN) * 3;
        c100 = {p[0], p[1], p[2]};  c101 = {p[3], p[4], p[5]};
        p = lut + (size_t)(e + LUTN * LUTN + LUTN) * 3;
        c110 = {p[0], p[1], p[2]};  c111 = {p[3], p[4], p[5]};
    }
    // Reference order: r, then g, then b; a*(1-f)+b*f form.
    F3 c00 = mix3(c000, c100, tr);
    F3 c01 = mix3(c001, c101, tr);
    F3 c10 = mix3(c010, c110, tr);
    F3 c11 = mix3(c011, c111, tr);
    F3 c0  = mix3(c00, c10, tg);
    F3 c1  = mix3(c01, c11, tg);
    return mix3(c0, c1, tb);
}

// ---- LUT repack: (33,33,33,3) f32 -> padded float4 ------------------------
__global__ void __launch_bounds__(256)
pack_lut_kernel(const float* __restrict__ lut, v4f* __restrict__ lutp) {
    int i = blockIdx.x * 256 + threadIdx.x;
    if (i < LUTENT) {
        const float* p = lut + (size_t)i * 3;
        v4f q = {p[0], p[1], p[2], 0.0f};
        lutp[i] = q;
    }
}

// ---- main kernel -----------------------------------------------------------
template <bool PADDED>
__global__ void __launch_bounds__(BLOCK)
lut3d_kernel(const float* __restrict__ img,
             const float* __restrict__ lut,
             const v4f* __restrict__ lutp,
             float* __restrict__ out) {
    __shared__ v4f sR[2][BLOCK];
    __shared__ v4f sG[2][BLOCK];
    __shared__ v4f sB[2][BLOCK];

    const unsigned tid   = threadIdx.x;
    const unsigned g0    = blockIdx.x * (unsigned)(BLOCK * ITER);
    const unsigned batch = g0 / PLANE_GRP;          // whole block in one batch
    const unsigned wg0   = g0 % PLANE_GRP;

    const v4f* R = (const v4f*)img + (size_t)batch * 3 * PLANE_GRP + wg0;
    const v4f* G = R + PLANE_GRP;
    const v4f* B = R + 2 * PLANE_GRP;
    v4f* oR = (v4f*)out + (size_t)batch * 3 * PLANE_GRP + wg0;
    v4f* oG = oR + PLANE_GRP;
    v4f* oB = oR + 2 * PLANE_GRP;

    // Prologue: async-stage chunk 0.
    {
        unsigned idx = tid;
        async_copy16(R + idx, &sR[0][tid]);
        async_copy16(G + idx, &sG[0][tid]);
        async_copy16(B + idx, &sB[0][tid]);
    }

    for (int k = 0; k < ITER; ++k) {
        const int cur = k & 1;
        if (k + 1 < ITER) {
            unsigned idx = (unsigned)(k + 1) * BLOCK + tid;
            async_copy16(R + idx, &sR[cur ^ 1][tid]);
            async_copy16(G + idx, &sG[cur ^ 1][tid]);
            async_copy16(B + idx, &sB[cur ^ 1][tid]);
            wait_async_le3();   // chunk k's 3 loads are the oldest -> done
        } else {
            wait_async_0();
        }

        const v4f r4 = sR[cur][tid];
        const v4f g4 = sG[cur][tid];
        const v4f b4 = sB[cur][tid];

        F3 o0 = sample_lut<PADDED>(lutp, lut, r4.x, g4.x, b4.x);
        F3 o1 = sample_lut<PADDED>(lutp, lut, r4.y, g4.y, b4.y);
        F3 o2 = sample_lut<PADDED>(lutp, lut, r4.z, g4.z, b4.z);
        F3 o3 = sample_lut<PADDED>(lutp, lut, r4.w, g4.w, b4.w);

        v4f outR = {o0.x, o1.x, o2.x, o3.x};
        v4f outG = {o0.y, o1.y, o2.y, o3.y};
        v4f outB = {o0.z, o1.z, o2.z, o3.z};

        const unsigned idx = (unsigned)k * BLOCK + tid;
        __builtin_nontemporal_store(outR, oR + idx);
        __builtin_nontemporal_store(outG, oG + idx);
        __builtin_nontemporal_store(outB, oB + idx);
    }
}

// ---------------------------------------------------------------------------
extern "C" void kernel_launch(void* const* d_in, const int* in_sizes, int n_in,
                              void* d_out, int out_size, void* d_ws, size_t ws_size,
                              hipStream_t stream) {
    const float* img = (const float*)d_in[0];   // (4,3,2048,2048) f32
    const float* lut = (const float*)d_in[1];   // (33,33,33,3) f32
    float* out = (float*)d_out;

    const unsigned totalPixels = (unsigned)(in_sizes[0] / 3);   // 16,777,216
    const unsigned totalGroups = totalPixels / 4;               // 4,194,304
    const unsigned blocks      = totalGroups / (BLOCK * ITER);  // 2048

    const size_t padBytes = (size_t)LUTENT * sizeof(v4f);       // ~575 KB
    if (ws_size >= padBytes) {
        v4f* lutp = (v4f*)d_ws;
        pack_lut_kernel<<<(LUTENT + 255) / 256, 256, 0, stream>>>(lut, lutp);
        lut3d_kernel<true><<<blocks, BLOCK, 0, stream>>>(img, lut, lutp, out);
    } else {
        lut3d_kernel<false><<<blocks, BLOCK, 0, stream>>>(img, lut, nullptr, out);
    }
}